// GVPConvLayer_9148280340760
// MI455X (gfx1250) — compile-verified
//
#include <hip/hip_runtime.h>
#include <hip/hip_bf16.h>

#define N_NODES 50000
#define N_EDGES 800000
#define NS 100
#define NV 16
#define ES 32
#define EPSF 1e-8f
#define LN_EPSF 1e-5f

typedef float v2f __attribute__((ext_vector_type(2)));
typedef float v8f __attribute__((ext_vector_type(8)));

struct PackedGvp { const float *wh, *ws, *bs, *wv; };

// ---------------------------------------------------------------------------
// Weight packing: fragment f = (t*NK + ks) holds, for lane l, the B-matrix
// pair B[kk..kk+1][n] = W[n][kk..kk+1] with n = t*16 + (l&15),
// kk = ks*4 + 2*(l>>4), zero-padded outside [SO) x [Kw).
// Consumers then do a single unconditional 8B load per (tile, kstep).
// ---------------------------------------------------------------------------
__global__ __launch_bounds__(256) void pack_w_kernel(
    const float* __restrict__ W, float* __restrict__ dst, int SO, int Kw,
    int NT, int NK)
{
  int gid = blockIdx.x * 256 + threadIdx.x;
  int total = NT * NK * 32;
  if (gid >= total) return;
  int l = gid & 31, f = gid >> 5;
  int t = f / NK, ks = f - t * NK;
  int n  = (t << 4) + (l & 15);
  int kk = (ks << 2) + ((l >> 4) << 1);
  float x = (n < SO && kk     < Kw) ? W[(size_t)n * Kw + kk]     : 0.f;
  float y = (n < SO && kk + 1 < Kw) ? W[(size_t)n * Kw + kk + 1] : 0.f;
  dst[(size_t)f * 64 + l * 2 + 0] = x;
  dst[(size_t)f * 64 + l * 2 + 1] = y;
}

__global__ __launch_bounds__(256) void pack_bias_kernel(
    const float* __restrict__ b, float* __restrict__ dst, int SO, int tot)
{
  int i = blockIdx.x * 256 + threadIdx.x;
  if (i < tot) dst[i] = (i < SO) ? b[i] : 0.f;
}

// ---------------------------------------------------------------------------
// D[16][NT*16] = A[16][4*nk] * packed(W)^T (+biasPad, +relu).  One wave32.
// Up to 4 independent accumulator chains per chunk; A loaded once per kstep.
// ---------------------------------------------------------------------------
template<int NT, bool RELU>
__device__ __forceinline__ void linf(
    const float* __restrict__ A, int lda, int nk,
    const float* __restrict__ Wp, const float* __restrict__ biasPad,
    float* __restrict__ out, int ldo, int lane)
{
  const int m     = lane & 15;
  const int koff  = (lane >> 4) << 1;
  const int rbase = (lane >> 4) << 3;
  const float* Arow = A + m * lda + koff;
#pragma unroll
  for (int t0 = 0; t0 < NT; t0 += 4) {
    v8f acc[4];
#pragma unroll
    for (int u = 0; u < 4; ++u) acc[u] = (v8f){0.f,0.f,0.f,0.f,0.f,0.f,0.f,0.f};
    const float* p = Wp + (size_t)t0 * nk * 64 + (lane << 1);
#pragma unroll 2
    for (int ks = 0; ks < nk; ++ks) {
      v2f a;
      a.x = Arow[ks * 4];
      a.y = Arow[ks * 4 + 1];
#pragma unroll
      for (int u = 0; u < 4; ++u) {
        if (t0 + u < NT) {
          v2f b = *(const v2f*)(p + ((size_t)u * nk + ks) * 64);
          acc[u] = __builtin_amdgcn_wmma_f32_16x16x4_f32(
              false, a, false, b, (short)0, acc[u], false, false);
        }
      }
    }
#pragma unroll
    for (int u = 0; u < 4; ++u) {
      if (t0 + u < NT) {
        const int n = (t0 + u) * 16 + m;
        const float bv = biasPad ? biasPad[n] : 0.f;
#pragma unroll
        for (int r = 0; r < 8; ++r) {
          float val = acc[u][r] + bv;
          if (RELU) val = fmaxf(val, 0.f);
          out[(rbase + r) * ldo + n] = val;
        }
      }
    }
  }
}

// One GVP layer for a 16-row tile. sbuf cols [0,SI) = s-input; vn written at
// [SI,SI+H); cols [SI+H,KS4) zeroed. vin planes zero-padded to round4(VI).
template<int SI, int VI, int SO, int VO, int H, bool ACT>
__device__ void gvp16(float* sbuf, int lds_s, PackedGvp w,
                      float* vi0, float* vi1, float* vi2, int lds_vi,
                      float* vh0, float* vh1, float* vh2, int lds_vh,
                      float* sout, int lds_so,
                      float* vo0, float* vo1, float* vo2, int lds_vo,
                      int lane)
{
  constexpr int VI4 = (VI + 3) & ~3;
  constexpr int H4  = (H + 3) & ~3;
  constexpr int KS  = SI + H;
  constexpr int KS4 = (KS + 3) & ~3;
  constexpr int HNT = (H + 15) >> 4;
  constexpr int SNT = (SO + 15) >> 4;
  constexpr int VNT = (VO + 15) >> 4;

  // vh = v . wh^T per coordinate (cols >= H come out exactly 0)
  linf<HNT, false>(vi0, lds_vi, VI4 / 4, w.wh, nullptr, vh0, lds_vh, lane);
  linf<HNT, false>(vi1, lds_vi, VI4 / 4, w.wh, nullptr, vh1, lds_vh, lane);
  linf<HNT, false>(vi2, lds_vi, VI4 / 4, w.wh, nullptr, vh2, lds_vh, lane);
  __syncthreads();
  // vn -> sbuf[:, SI..SI+H)
  for (int idx = lane; idx < 16 * H; idx += 32) {
    int e = idx / H, j = idx - e * H;
    float x = vh0[e * lds_vh + j], y = vh1[e * lds_vh + j], z = vh2[e * lds_vh + j];
    sbuf[e * lds_s + SI + j] = sqrtf(fmaxf(x * x + y * y + z * z, EPSF));
  }
  if constexpr (KS4 > KS) {
    for (int idx = lane; idx < 16 * (KS4 - KS); idx += 32) {
      int e = idx / (KS4 - KS), j = idx - e * (KS4 - KS);
      sbuf[e * lds_s + KS + j] = 0.f;
    }
  }
  __syncthreads();
  // s' = [s || vn] . ws^T + b (relu if ACT)
  linf<SNT, ACT>(sbuf, lds_s, KS4 / 4, w.ws, w.bs, sout, lds_so, lane);
  // vo = vh . wv^T per coordinate
  linf<VNT, false>(vh0, lds_vh, H4 / 4, w.wv, nullptr, vo0, lds_vo, lane);
  linf<VNT, false>(vh1, lds_vh, H4 / 4, w.wv, nullptr, vo1, lds_vo, lane);
  linf<VNT, false>(vh2, lds_vh, H4 / 4, w.wv, nullptr, vo2, lds_vo, lane);
  __syncthreads();
  if (ACT) {
    for (int idx = lane; idx < 16 * VO; idx += 32) {
      int e = idx / VO, o = idx - e * VO;
      float x = vo0[e * lds_vo + o], y = vo1[e * lds_vo + o], z = vo2[e * lds_vo + o];
      float nr = sqrtf(fmaxf(x * x + y * y + z * z, EPSF));
      float g = 1.f / (1.f + __expf(-nr));
      vo0[e * lds_vo + o] = x * g;
      vo1[e * lds_vo + o] = y * g;
      vo2[e * lds_vo + o] = z * g;
    }
    __syncthreads();
  }
}

// ---------------- Edge message kernel: one wave per 16 edges ----------------
__global__ __launch_bounds__(32) void edge_msg_kernel(
    const float* __restrict__ s, const float* __restrict__ v,
    const int* __restrict__ eidx,
    const float* __restrict__ edge_s, const float* __restrict__ edge_v,
    PackedGvp m0, PackedGvp m1, PackedGvp m2,
    float* __restrict__ agg_s, float* __restrict__ agg_v, float* __restrict__ cnt)
{
  __shared__ __align__(16) float sA[16 * 272];   // msg0 s-input (265 -> 268 pad)
  __shared__ __align__(16) float sB[16 * 116];   // ping-pong s
  __shared__ __align__(16) float VA[3 * 16 * 48];
  __shared__ __align__(16) float VB[3 * 16 * 48];
  __shared__ __align__(16) float VC[3 * 16 * 48];
  float* A0 = VA; float* A1 = VA + 768; float* A2 = VA + 1536;
  float* B0 = VB; float* B1 = VB + 768; float* B2 = VB + 1536;
  float* C0 = VC; float* C1 = VC + 768; float* C2 = VC + 1536;

  const int lane = threadIdx.x;
  const int e  = lane & 15;
  const int hf = lane >> 4;
  const long ge = (long)blockIdx.x * 16 + e;
  const int src = eidx[ge];
  const int dst = eidx[N_EDGES + ge];

  // gather scalars as float4: [s[src](25 f4) | edge_s(8 f4) | s[dst](25 f4)]
  {
    const float4* s4  = (const float4*)s;
    const float4* es4 = (const float4*)edge_s;
    float4* row4 = (float4*)&sA[e * 272];
    for (int c = hf * 29; c < hf * 29 + 29; ++c) {
      float4 val;
      if (c < 25)      val = s4[(size_t)src * 25 + c];
      else if (c < 33) val = es4[(size_t)ge * 8 + (c - 25)];
      else             val = s4[(size_t)dst * 25 + (c - 33)];
      row4[c] = val;   // dest float4 index == c (100/4=25, 132/4=33)
    }
  }
  // gather vectors: rows 0..15 v[src], 16 edge_v, 17..32 v[dst]; 33..35 zero
  for (int j = hf; j < 36; j += 2) {
    float x = 0.f, y = 0.f, z = 0.f;
    if (j < NV)            { const float* p = v + ((long)src * NV + j) * 3;            x = p[0]; y = p[1]; z = p[2]; }
    else if (j == NV)      { const float* p = edge_v + ge * 3;                         x = p[0]; y = p[1]; z = p[2]; }
    else if (j < 2*NV + 1) { const float* p = v + ((long)dst * NV + (j - NV - 1)) * 3; x = p[0]; y = p[1]; z = p[2]; }
    A0[e * 48 + j] = x; A1[e * 48 + j] = y; A2[e * 48 + j] = z;
  }
  __syncthreads();

  gvp16<232, 33, 100, 16, 33, true >(sA, 272, m0, A0, A1, A2, 48, B0, B1, B2, 48,
                                     sB, 116, C0, C1, C2, 48, lane);
  gvp16<100, 16, 100, 16, 16, true >(sB, 116, m1, C0, C1, C2, 48, B0, B1, B2, 48,
                                     sA, 272, A0, A1, A2, 48, lane);
  gvp16<100, 16, 100, 16, 16, false>(sA, 272, m2, A0, A1, A2, 48, B0, B1, B2, 48,
                                     sB, 116, C0, C1, C2, 48, lane);
  __syncthreads();

  // scatter-add messages into node accumulators (L2-resident fp32 atomics)
  for (int i = hf * 50; i < hf * 50 + 50; ++i)
    atomicAdd(&agg_s[(long)dst * NS + i], sB[e * 116 + i]);
  float* Cp[3] = {C0, C1, C2};
  for (int i = hf * 24; i < hf * 24 + 24; ++i) {
    int j = i / 3, c = i - j * 3;
    atomicAdd(&agg_v[(long)dst * (NV * 3) + i], Cp[c][e * 48 + j]);
  }
  if (hf == 0) atomicAdd(&cnt[dst], 1.0f);
}

// ---------------- Node update kernel: one wave per 16 nodes ----------------
__global__ __launch_bounds__(32) void node_update_kernel(
    const float* __restrict__ s, const float* __restrict__ v,
    const float* __restrict__ agg_s, const float* __restrict__ agg_v,
    const float* __restrict__ cnt,
    PackedGvp f0, PackedGvp f1,
    const float* __restrict__ ln0g, const float* __restrict__ ln0b,
    const float* __restrict__ ln1g, const float* __restrict__ ln1b,
    float* __restrict__ out)
{
  __shared__ __align__(16) float SB1[16 * 132];   // s1 (post-LN0) + ff0 vn
  __shared__ __align__(16) float SBIG[16 * 432];  // ff0 s-out (400) + ff1 vn
  __shared__ __align__(16) float SO2[16 * 112];   // ff1 s-out
  __shared__ __align__(16) float VA[3 * 16 * 32], VB[3 * 16 * 32], VC[3 * 16 * 32];
  __shared__ __align__(16) float VD[3 * 16 * 16];
  float* A0 = VA; float* A1 = VA + 512; float* A2 = VA + 1024;
  float* B0 = VB; float* B1 = VB + 512; float* B2 = VB + 1024;
  float* C0 = VC; float* C1 = VC + 512; float* C2 = VC + 1024;
  float* D0 = VD; float* D1 = VD + 256; float* D2 = VD + 512;

  const int lane = threadIdx.x;
  const int e  = lane & 15;
  const int hf = lane >> 4;
  const long node = (long)blockIdx.x * 16 + e;

  const float inv = 1.f / fmaxf(cnt[node], 1.f);
  for (int i = hf * 50; i < hf * 50 + 50; ++i)
    SB1[e * 132 + i] = s[node * NS + i] + agg_s[node * NS + i] * inv;
  float* Ap[3] = {A0, A1, A2};
  for (int i = hf * 24; i < hf * 24 + 24; ++i) {
    int j = i / 3, c = i - j * 3;
    Ap[c][e * 32 + j] = v[node * (NV * 3) + i] + agg_v[node * (NV * 3) + i] * inv;
  }
  __syncthreads();

  // lanes 0..15: LayerNorm0 in place; lanes 16..31: vnorm in place
  if (hf == 0) {
    float mu = 0.f;
    for (int i = 0; i < NS; ++i) mu += SB1[e * 132 + i];
    mu *= (1.f / NS);
    float var = 0.f;
    for (int i = 0; i < NS; ++i) { float d = SB1[e * 132 + i] - mu; var += d * d; }
    float rstd = rsqrtf(var * (1.f / NS) + LN_EPSF);
    for (int i = 0; i < NS; ++i)
      SB1[e * 132 + i] = (SB1[e * 132 + i] - mu) * rstd * ln0g[i] + ln0b[i];
  } else {
    float acc = 0.f;
    for (int j = 0; j < NV; ++j) {
      float x = A0[e*32+j], y = A1[e*32+j], z = A2[e*32+j];
      acc += fmaxf(x * x + y * y + z * z, EPSF);
    }
    float ivn = 1.f / sqrtf(acc * (1.f / NV));
    for (int j = 0; j < NV; ++j) { A0[e*32+j] *= ivn; A1[e*32+j] *= ivn; A2[e*32+j] *= ivn; }
  }
  __syncthreads();

  gvp16<100, 16, 400, 32, 32, true >(SB1, 132, f0, A0, A1, A2, 32, B0, B1, B2, 32,
                                     SBIG, 432, C0, C1, C2, 32, lane);
  gvp16<400, 32, 100, 16, 32, false>(SBIG, 432, f1, C0, C1, C2, 32, B0, B1, B2, 32,
                                     SO2, 112, D0, D1, D2, 16, lane);
  __syncthreads();

  // s2 = LN1(s1 + fs); v2 = vnorm(v1 + fv)  (v1 = VA, fv = VD)
  if (hf == 0) {
    float mu = 0.f;
    for (int i = 0; i < NS; ++i) mu += SB1[e * 132 + i] + SO2[e * 112 + i];
    mu *= (1.f / NS);
    float var = 0.f;
    for (int i = 0; i < NS; ++i) {
      float d = SB1[e * 132 + i] + SO2[e * 112 + i] - mu; var += d * d;
    }
    float rstd = rsqrtf(var * (1.f / NS) + LN_EPSF);
    for (int i = 0; i < NS; ++i) {
      float val = (SB1[e * 132 + i] + SO2[e * 112 + i] - mu) * rstd * ln1g[i] + ln1b[i];
      out[node * NS + i] = val;
    }
  } else {
    float acc = 0.f;
    for (int j = 0; j < NV; ++j) {
      float x = A0[e*32+j] + D0[e*16+j];
      float y = A1[e*32+j] + D1[e*16+j];
      float z = A2[e*32+j] + D2[e*16+j];
      acc += fmaxf(x * x + y * y + z * z, EPSF);
    }
    float ivn = 1.f / sqrtf(acc * (1.f / NV));
    float* Dp[3] = {D0, D1, D2};
    for (int j = 0; j < NV; ++j)
      for (int c = 0; c < 3; ++c)
        out[(long)N_NODES * NS + node * (NV * 3) + j * 3 + c] =
            (Ap[c][e * 32 + j] + Dp[c][e * 16 + j]) * ivn;
  }
}

// ---------------------------------------------------------------------------
static inline size_t frag_floats(int SO, int Kw) {
  return (size_t)((SO + 15) / 16) * ((Kw + 3) / 4) * 64;
}

extern "C" void kernel_launch(void* const* d_in, const int* in_sizes, int n_in,
                              void* d_out, int out_size, void* d_ws, size_t ws_size,
                              hipStream_t stream)
{
  const float* s      = (const float*)d_in[0];
  const float* v      = (const float*)d_in[1];
  const int*   eidx   = (const int*)d_in[2];
  const float* edge_s = (const float*)d_in[3];
  const float* edge_v = (const float*)d_in[4];
  const float* W[5][4];   // [gvp][wh, ws_w, ws_b, wv]
  for (int g = 0; g < 5; ++g)
    for (int k = 0; k < 4; ++k)
      W[g][k] = (const float*)d_in[5 + g * 4 + k];
  const float* ln0g = (const float*)d_in[25];
  const float* ln0b = (const float*)d_in[26];
  const float* ln1g = (const float*)d_in[27];
  const float* ln1b = (const float*)d_in[28];

  float* ws = (float*)d_ws;
  float* agg_s = ws;                                    // [N,100]
  float* agg_v = agg_s + (size_t)N_NODES * NS;          // [N,16,3]
  float* cntp  = agg_v + (size_t)N_NODES * NV * 3;      // [N]
  size_t off = (size_t)N_NODES * (NS + NV * 3 + 1);
  hipMemsetAsync(d_ws, 0, off * sizeof(float), stream);

  // GVP dims: {SO_wh(=H), Kw_wh(=VI), SO_ws, Kw_ws(=SI+H), SO_wv(=VO), Kw_wv(=H)}
  static const int dims[5][6] = {
      {33, 33, 100, 265, 16, 33},   // msg0
      {16, 16, 100, 116, 16, 16},   // msg1
      {16, 16, 100, 116, 16, 16},   // msg2
      {32, 16, 400, 132, 32, 32},   // ff0
      {32, 32, 100, 432, 16, 32},   // ff1
  };
  PackedGvp P[5];
  for (int g = 0; g < 5; ++g) {
    const float* pw[3];
    int mats[3][2] = {{dims[g][0], dims[g][1]},
                      {dims[g][2], dims[g][3]},
                      {dims[g][4], dims[g][5]}};
    const float* src[3] = {W[g][0], W[g][1], W[g][3]};
    for (int mIdx = 0; mIdx < 3; ++mIdx) {
      int SO = mats[mIdx][0], Kw = mats[mIdx][1];
      int NT = (SO + 15) / 16, NK = (Kw + 3) / 4;
      int total = NT * NK * 32;
      pack_w_kernel<<<(total + 255) / 256, 256, 0, stream>>>(
          src[mIdx], ws + off, SO, Kw, NT, NK);
      pw[mIdx] = ws + off;
      off += frag_floats(SO, Kw);
    }
    // padded bias
    int SOb = dims[g][2];
    int btot = ((SOb + 15) / 16) * 16;
    pack_bias_kernel<<<(btot + 255) / 256, 256, 0, stream>>>(
        W[g][2], ws + off, SOb, btot);
    P[g] = {pw[0], pw[1], ws + off, pw[2]};
    off += (size_t)btot;
  }

  edge_msg_kernel<<<N_EDGES / 16, 32, 0, stream>>>(
      s, v, eidx, edge_s, edge_v, P[0], P[1], P[2], agg_s, agg_v, cntp);
  node_update_kernel<<<N_NODES / 16, 32, 0, stream>>>(
      s, v, agg_s, agg_v, cntp, P[3], P[4], ln0g, ln0b, ln1g, ln1b, (float*)d_out);
}